// GCN_38491496907445
// MI455X (gfx1250) — compile-verified
//
#include <hip/hip_runtime.h>
#include <hip/hip_bf16.h>
#include <math.h>

// ---------------------------------------------------------------------------
// GCN forward for MI455X (gfx1250, wave32).
// GEMMs use V_WMMA_F32_16X16X4_F32 (full f32 precision matrix cores).
// Edge aggregation is atomic-add scatter (bandwidth bound: ~2.5 GB total,
// ~100us at 23.3 TB/s -- this dominates; GEMMs are ~2.8 GFLOP, negligible).
// ---------------------------------------------------------------------------

typedef __attribute__((ext_vector_type(2))) float v2f;
typedef __attribute__((ext_vector_type(8))) float v8f;

#define HIDDEN 64
#define NGRAPHS 1024
#define BIOPRINT 2048

// ---------------------------------------------------------------------------
// Generic WMMA f32 GEMM:  C[M x N] = op(A)[M x K] @ B[K x N] (+ bias)
// op(A) = relu(A) if reluIn.  Row-major everything.  M % 16 == 0, K % 4 == 0,
// N % 16 == 0.  Grid: (M/16, ceil(N/64)), block 128 (4 waves, 1 tile/wave).
// ---------------------------------------------------------------------------
__global__ void gcn_wmma_gemm(const float* __restrict__ A,
                              const float* __restrict__ B,
                              const float* __restrict__ bias,
                              float* __restrict__ C,
                              int N, int K, int reluIn) {
    const int wave = threadIdx.x >> 5;
    const int lane = threadIdx.x & 31;
    const int colBase = (blockIdx.y * 4 + wave) * 16;
    if (colBase >= N) return;  // uniform per wave: EXEC stays all-ones

    const int rowBase = blockIdx.x * 16;
    const int half = lane >> 4;   // which K-pair this lane group carries
    const int mr   = lane & 15;   // row (for A) / col (for B) within tile

    v8f acc = {0.f, 0.f, 0.f, 0.f, 0.f, 0.f, 0.f, 0.f};

    const float* arow = A + (size_t)(rowBase + mr) * K;
    for (int k = 0; k < K; k += 4) {
        const int ka = k + half * 2;
        // A 16x4 layout: lanes 0-15 hold K={k,k+1}, lanes 16-31 K={k+2,k+3}
        v2f av = *(const v2f*)(arow + ka);
        if (reluIn) { av.x = fmaxf(av.x, 0.f); av.y = fmaxf(av.y, 0.f); }
        // B 4x16 layout mirrors A: VGPR0 = row ka, VGPR1 = row ka+1 per group
        v2f bv;
        bv.x = B[(size_t)(ka + 0) * N + colBase + mr];
        bv.y = B[(size_t)(ka + 1) * N + colBase + mr];
        acc = __builtin_amdgcn_wmma_f32_16x16x4_f32(
            /*neg_a=*/false, av, /*neg_b=*/false, bv,
            /*c_mod=*/(short)0, acc, /*reuse_a=*/false, /*reuse_b=*/false);
    }

    // C/D layout: VGPR r -> row (r + 8*half), col = lane&15
    const int col = colBase + mr;
    const float badd = bias ? bias[col] : 0.f;
#pragma unroll
    for (int r = 0; r < 8; ++r) {
        const int row = rowBase + r + half * 8;
        C[(size_t)row * N + col] = acc[r] + badd;
    }
}

// ---------------------------------------------------------------------------
__global__ void gcn_zero(float* __restrict__ p, int n) {
    int i = blockIdx.x * blockDim.x + threadIdx.x;
    if (i < n) p[i] = 0.f;
}

__global__ void gcn_degree(const int* __restrict__ dst,
                           float* __restrict__ deg, int nEdges) {
    int e = blockIdx.x * blockDim.x + threadIdx.x;
    if (e < nEdges) atomicAdd(&deg[dst[e]], 1.0f);
}

__global__ void gcn_dinv(float* __restrict__ deg, int nNodes) {
    int i = blockIdx.x * blockDim.x + threadIdx.x;
    if (i < nNodes) deg[i] = rsqrtf(deg[i] + 1.0f);  // deg+self-loop >= 1
}

// agg = hw * dinv^2 + b   (self-loop term + bias as aggregation init)
__global__ void gcn_self_init(const float* __restrict__ hw,
                              const float* __restrict__ dinv,
                              const float* __restrict__ b,
                              float* __restrict__ agg, int nNodes) {
    int t = blockIdx.x * blockDim.x + threadIdx.x;
    if (t < nNodes * HIDDEN) {
        int i = t >> 6;
        int f = t & 63;
        float dv = dinv[i];
        agg[t] = hw[t] * dv * dv + b[f];
    }
}

// agg[dst] += hw[src] * dinv[src]*dinv[dst]  (64 threads per edge)
__global__ void gcn_scatter(const float* __restrict__ hw,
                            const float* __restrict__ dinv,
                            const int* __restrict__ src,
                            const int* __restrict__ dst,
                            float* __restrict__ agg, int nEdges) {
    long long t = (long long)blockIdx.x * blockDim.x + threadIdx.x;
    if (t < (long long)nEdges * HIDDEN) {
        int e = (int)(t >> 6);
        int f = (int)(t & 63);
        int s = src[e];
        int d = dst[e];
        float w = dinv[s] * dinv[d];
        atomicAdd(&agg[(size_t)d * HIDDEN + f], hw[(size_t)s * HIDDEN + f] * w);
    }
}

__global__ void gcn_pool(const float* __restrict__ h,
                         const int* __restrict__ batch,
                         float* __restrict__ pooled,
                         float* __restrict__ cnt, int nNodes) {
    int t = blockIdx.x * blockDim.x + threadIdx.x;
    if (t < nNodes * HIDDEN) {
        int i = t >> 6;
        int f = t & 63;
        int g = batch[i];
        atomicAdd(&pooled[(size_t)g * HIDDEN + f], h[t]);
        if (f == 0) atomicAdd(&cnt[g], 1.0f);
    }
}

__global__ void gcn_pool_div(float* __restrict__ pooled,
                             const float* __restrict__ cnt) {
    int t = blockIdx.x * blockDim.x + threadIdx.x;
    if (t < NGRAPHS * HIDDEN) pooled[t] /= fmaxf(cnt[t >> 6], 1.0f);
}

// Per-row softmax over 2048 logits, then threshold >= 0.5 -> {0,1} floats.
__global__ void gcn_softmax_thresh(const float* __restrict__ logits,
                                   float* __restrict__ out) {
    __shared__ float red[256];
    const int row = blockIdx.x;
    const float* l = logits + (size_t)row * BIOPRINT;
    float m = -INFINITY;
    for (int c = threadIdx.x; c < BIOPRINT; c += 256) m = fmaxf(m, l[c]);
    red[threadIdx.x] = m;
    __syncthreads();
    for (int s = 128; s > 0; s >>= 1) {
        if (threadIdx.x < s)
            red[threadIdx.x] = fmaxf(red[threadIdx.x], red[threadIdx.x + s]);
        __syncthreads();
    }
    m = red[0];
    __syncthreads();
    float sum = 0.f;
    for (int c = threadIdx.x; c < BIOPRINT; c += 256) sum += expf(l[c] - m);
    red[threadIdx.x] = sum;
    __syncthreads();
    for (int s = 128; s > 0; s >>= 1) {
        if (threadIdx.x < s) red[threadIdx.x] += red[threadIdx.x + s];
        __syncthreads();
    }
    const float inv = 1.0f / red[0];
    for (int c = threadIdx.x; c < BIOPRINT; c += 256)
        out[(size_t)row * BIOPRINT + c] =
            (expf(l[c] - m) * inv >= 0.5f) ? 1.0f : 0.0f;
}

// ---------------------------------------------------------------------------
extern "C" void kernel_launch(void* const* d_in, const int* in_sizes, int n_in,
                              void* d_out, int out_size, void* d_ws, size_t ws_size,
                              hipStream_t stream) {
    const float* x   = (const float*)d_in[0];
    const int* eidx  = (const int*)d_in[1];     // [2, E] flat: src then dst
    const int* batch = (const int*)d_in[2];
    const float* W1  = (const float*)d_in[3];
    const float* b1  = (const float*)d_in[4];
    const float* W2  = (const float*)d_in[5];
    const float* b2  = (const float*)d_in[6];
    const float* W3  = (const float*)d_in[7];
    const float* b3  = (const float*)d_in[8];
    const float* Wd  = (const float*)d_in[9];
    const float* bd  = (const float*)d_in[10];
    float* out = (float*)d_out;

    const int nNodes = in_sizes[0] / HIDDEN;     // 100000 (multiple of 16)
    const int nEdges = in_sizes[1] / 2;          // 1.6M
    const int* src = eidx;
    const int* dst = eidx + nEdges;

    // Workspace layout (floats)
    float* ws = (float*)d_ws;
    float* dinv   = ws;                                  // nNodes (rounded)
    float* bufA   = dinv + 102400;                       // nNodes*64
    float* bufB   = bufA + (size_t)nNodes * HIDDEN;      // nNodes*64
    float* pooled = bufB + (size_t)nNodes * HIDDEN;      // 1024*64
    float* cnt    = pooled + NGRAPHS * HIDDEN;           // 1024
    float* logits = cnt + NGRAPHS;                       // 1024*2048

    const int T = 256;
    const long long nf = (long long)nNodes * HIDDEN;
    const long long ef = (long long)nEdges * HIDDEN;

    // Degree -> dinv
    gcn_zero<<<(nNodes + T - 1) / T, T, 0, stream>>>(dinv, nNodes);
    gcn_degree<<<(nEdges + T - 1) / T, T, 0, stream>>>(dst, dinv, nEdges);
    gcn_dinv<<<(nNodes + T - 1) / T, T, 0, stream>>>(dinv, nNodes);

    const dim3 gemmGridN(nNodes / 16, 1);   // N=64 -> 4 waves cover all cols

    // Layer 1: hw = x@W1 (bufA); agg = init+scatter (bufB)
    gcn_wmma_gemm<<<gemmGridN, 128, 0, stream>>>(x, W1, nullptr, bufA, HIDDEN, HIDDEN, 0);
    gcn_self_init<<<(nf + T - 1) / T, T, 0, stream>>>(bufA, dinv, b1, bufB, nNodes);
    gcn_scatter<<<(unsigned)((ef + T - 1) / T), T, 0, stream>>>(bufA, dinv, src, dst, bufB, nEdges);

    // Layer 2 (relu fused into A-load)
    gcn_wmma_gemm<<<gemmGridN, 128, 0, stream>>>(bufB, W2, nullptr, bufA, HIDDEN, HIDDEN, 1);
    gcn_self_init<<<(nf + T - 1) / T, T, 0, stream>>>(bufA, dinv, b2, bufB, nNodes);
    gcn_scatter<<<(unsigned)((ef + T - 1) / T), T, 0, stream>>>(bufA, dinv, src, dst, bufB, nEdges);

    // Layer 3 (relu fused, no relu on output)
    gcn_wmma_gemm<<<gemmGridN, 128, 0, stream>>>(bufB, W3, nullptr, bufA, HIDDEN, HIDDEN, 1);
    gcn_self_init<<<(nf + T - 1) / T, T, 0, stream>>>(bufA, dinv, b3, bufB, nNodes);
    gcn_scatter<<<(unsigned)((ef + T - 1) / T), T, 0, stream>>>(bufA, dinv, src, dst, bufB, nEdges);

    // Global mean pool
    gcn_zero<<<(NGRAPHS * HIDDEN + NGRAPHS + T - 1) / T, T, 0, stream>>>(pooled, NGRAPHS * HIDDEN + NGRAPHS);
    gcn_pool<<<(nf + T - 1) / T, T, 0, stream>>>(bufB, batch, pooled, cnt, nNodes);
    gcn_pool_div<<<(NGRAPHS * HIDDEN + T - 1) / T, T, 0, stream>>>(pooled, cnt);

    // Dense: logits = pooled @ Wd + bd   [1024x64]@[64x2048]
    const dim3 gemmGridD(NGRAPHS / 16, BIOPRINT / 64);
    gcn_wmma_gemm<<<gemmGridD, 128, 0, stream>>>(pooled, Wd, bd, logits, BIOPRINT, HIDDEN, 0);

    // Softmax + threshold
    gcn_softmax_thresh<<<NGRAPHS, 256, 0, stream>>>(logits, out);
}